// FractalGNNNode_83064667505144
// MI455X (gfx1250) — compile-verified
//
#include <hip/hip_runtime.h>

// ---------------------------------------------------------------------------
// FractalGNN step for MI455X (gfx1250, wave32).
//
// Roofline: ~318 MB of fp32 weights streamed per call vs ~160 MFLOP.
// => pure HBM-bandwidth bound (~13.7 us at 23.3 TB/s). Big GEMVs use
// wave-per-row coalesced float4 streaming (regular temporal hint so the
// 192 MB L2 retains most of agg_W1 across graph replays). The small MLP
// layers use V_WMMA_F32_16X16X4_F32 (full fp32 precision, compute is free).
// ---------------------------------------------------------------------------

#define S_DIM 64
#define C_DIM 27
#define SD    (S_DIM * 5)      // 320  per-node state size
#define KBIG  (C_DIM * SD)     // 8640 aggregator reduction length
#define HID   64               // hidden width of update nets
#define COMB  (SD * 2)         // 640  concat(state, agg)

typedef float v2f __attribute__((ext_vector_type(2)));
typedef float v4f __attribute__((ext_vector_type(4)));
typedef float v8f __attribute__((ext_vector_type(8)));

// ---------------------------------------------------------------------------
// Bandwidth kernel: out[r] = dot(W[r, 0:K], x[(r / rowsPerNode) * xNodeStride
//                                              : ... + K])
// One wave per row. Lanes read consecutive 16B chunks -> each wave issues
// perfectly coalesced 512B requests while streaming the row.
// K must be a multiple of 4 (K = 8640 here).
// ---------------------------------------------------------------------------
__global__ __launch_bounds__(256) void gemv_rows(
    const float* __restrict__ W, const float* __restrict__ x,
    float* __restrict__ out, int rowsTotal, int K,
    int rowsPerNode, int xNodeStride)
{
    const int lane = threadIdx.x & 31;
    const int wave = threadIdx.x >> 5;
    const int row  = blockIdx.x * 8 + wave;
    if (row >= rowsTotal) return;

    const float* __restrict__ wr = W + (size_t)row * (size_t)K;
    const float* __restrict__ xv = x + (size_t)(row / rowsPerNode) * (size_t)xNodeStride;

    float acc = 0.f;
    for (int j = lane * 4; j + 3 < K; j += 128) {
        v4f wv = *(const v4f*)(wr + j);
        v4f xw = *(const v4f*)(xv + j);
        acc += wv.x * xw.x + wv.y * xw.y + wv.z * xw.z + wv.w * xw.w;
    }
    // wave32 butterfly reduction
    #pragma unroll
    for (int off = 16; off > 0; off >>= 1)
        acc += __shfl_down(acc, off, 32);
    if (lane == 0) out[row] = acc;
}

// ---------------------------------------------------------------------------
// 16-row fp32 GEMV tile via V_WMMA_F32_16X16X4_F32.
// A (16x4): lane = M (lane&15), K = (lane>=16 ? 2 : 0) + vgpr  [ISA layout]
// B (4x16): activation chunk replicated into all 16 columns (same lane-half
//           K mapping as A), so every column of D accumulates the same dot.
// D (16x16 f32, 8 VGPRs): VGPR v, lanes 0-15 -> M=v, lanes 16-31 -> M=v+8.
// ---------------------------------------------------------------------------
__device__ inline v8f wmma_gemv16(const float* __restrict__ Wp,  // 16 x K, ld = K
                                  const float* __restrict__ xs,  // K (LDS)
                                  int K, int lane)
{
    v8f c = {0.f, 0.f, 0.f, 0.f, 0.f, 0.f, 0.f, 0.f};
    const int m    = lane & 15;
    const int koff = (lane >> 4) * 2;          // 0 or 2
    const float* __restrict__ wrow = Wp + (size_t)m * (size_t)K + koff;
    for (int k = 0; k < K; k += 4) {
        v2f a, b;
        a.x = wrow[k];
        a.y = wrow[k + 1];
        b.x = xs[k + koff];
        b.y = xs[k + koff + 1];
        c = __builtin_amdgcn_wmma_f32_16x16x4_f32(
                /*neg_a=*/false, a, /*neg_b=*/false, b,
                /*c_mod=*/(short)0, c, /*reuse_a=*/false, /*reuse_b=*/false);
    }
    return c;
}

// ---------------------------------------------------------------------------
// Level-1 update nets: one block (4 waves) per node.
//   comb = [states1[n], agg1[n] + agg_b1[n]]          (640, in LDS)
//   h    = relu(up_W1_1[n] @ comb + up_b1_1[n])       (64, WMMA: wave w -> rows 16w..16w+15)
//   new_states1[n] = states1[n] + up_W2_1[n] @ h + up_b2_1[n]
// ---------------------------------------------------------------------------
__global__ __launch_bounds__(128) void level1_mlp(
    const float* __restrict__ states1, const float* __restrict__ agg1raw,
    const float* __restrict__ agg_b1,
    const float* __restrict__ W1, const float* __restrict__ b1,
    const float* __restrict__ W2, const float* __restrict__ b2,
    float* __restrict__ newStates)
{
    __shared__ float comb[COMB];
    __shared__ float h[HID];
    const int n   = blockIdx.x;
    const int tid = threadIdx.x;

    for (int i = tid; i < SD; i += 128) {
        comb[i]      = states1[n * SD + i];
        comb[SD + i] = agg1raw[n * SD + i] + agg_b1[n * SD + i];
    }
    __syncthreads();

    const int lane = tid & 31;
    const int wave = tid >> 5;                     // 0..3 -> row tile
    v8f c = wmma_gemv16(W1 + (size_t)n * HID * COMB + (size_t)(wave * 16) * COMB,
                        comb, COMB, lane);
    if ((lane & 15) == 0) {                        // column-0 owners
        const int base = wave * 16 + (lane >> 4) * 8;
        #pragma unroll
        for (int v = 0; v < 8; ++v) {
            float hv = c[v] + b1[n * HID + base + v];
            h[base + v] = hv > 0.f ? hv : 0.f;
        }
    }
    __syncthreads();

    for (int i = tid; i < SD; i += 128) {
        const float* __restrict__ w2r = W2 + (size_t)n * SD * HID + (size_t)i * HID;
        float acc = b2[n * SD + i];
        #pragma unroll 8
        for (int k = 0; k < HID; ++k) acc += w2r[k] * h[k];
        newStates[n * SD + i] = states1[n * SD + i] + acc;
    }
}

// ---------------------------------------------------------------------------
// Root update net: single block (4 waves).
// ---------------------------------------------------------------------------
__global__ __launch_bounds__(128) void root_mlp(
    const float* __restrict__ state0, const float* __restrict__ extin,
    const float* __restrict__ agg0raw, const float* __restrict__ agg_b0,
    const float* __restrict__ W1, const float* __restrict__ b1,
    const float* __restrict__ W2, const float* __restrict__ b2,
    float* __restrict__ out)
{
    __shared__ float comb[COMB];
    __shared__ float h[HID];
    const int tid = threadIdx.x;

    for (int i = tid; i < SD; i += 128) {
        comb[i]      = state0[i] + extin[i];        // s0
        comb[SD + i] = agg0raw[i] + agg_b0[i];      // agg0
    }
    __syncthreads();

    const int lane = tid & 31;
    const int wave = tid >> 5;
    v8f c = wmma_gemv16(W1 + (size_t)(wave * 16) * COMB, comb, COMB, lane);
    if ((lane & 15) == 0) {
        const int base = wave * 16 + (lane >> 4) * 8;
        #pragma unroll
        for (int v = 0; v < 8; ++v) {
            float hv = c[v] + b1[base + v];
            h[base + v] = hv > 0.f ? hv : 0.f;
        }
    }
    __syncthreads();

    for (int i = tid; i < SD; i += 128) {
        const float* __restrict__ w2r = W2 + (size_t)i * HID;
        float acc = b2[i];
        #pragma unroll 8
        for (int k = 0; k < HID; ++k) acc += w2r[k] * h[k];
        out[i] = comb[i] + acc;                     // comb[i] still holds s0
    }
}

// ---------------------------------------------------------------------------
extern "C" void kernel_launch(void* const* d_in, const int* in_sizes, int n_in,
                              void* d_out, int out_size, void* d_ws, size_t ws_size,
                              hipStream_t stream)
{
    const float* extin   = (const float*)d_in[0];
    const float* state0  = (const float*)d_in[1];
    const float* states1 = (const float*)d_in[2];
    const float* states2 = (const float*)d_in[3];   // leaf_flat: [27][8640]
    const float* agg_W0  = (const float*)d_in[4];
    const float* agg_b0  = (const float*)d_in[5];
    const float* agg_W1  = (const float*)d_in[6];
    const float* agg_b1  = (const float*)d_in[7];
    const float* up_W1_0 = (const float*)d_in[8];
    const float* up_b1_0 = (const float*)d_in[9];
    const float* up_W2_0 = (const float*)d_in[10];
    const float* up_b2_0 = (const float*)d_in[11];
    const float* up_W1_1 = (const float*)d_in[12];
    const float* up_b1_1 = (const float*)d_in[13];
    const float* up_W2_1 = (const float*)d_in[14];
    const float* up_b2_1 = (const float*)d_in[15];

    float* out     = (float*)d_out;
    float* ws      = (float*)d_ws;
    float* agg1raw = ws;                  // 27*320 = 8640 floats
    float* newS1   = ws + C_DIM * SD;     // 27*320 = 8640 floats
    float* agg0raw = ws + 2 * C_DIM * SD; // 320 floats

    // 1) agg1[n] = agg_W1[n] @ leaf_flat[n]   (27*320 rows, K=8640) — the 298 MB stream
    gemv_rows<<<(C_DIM * SD) / 8, 256, 0, stream>>>(
        agg_W1, states2, agg1raw, C_DIM * SD, KBIG, SD, KBIG);

    // 2) per-node MLP + residual -> new_states1
    level1_mlp<<<C_DIM, 128, 0, stream>>>(
        states1, agg1raw, agg_b1, up_W1_1, up_b1_1, up_W2_1, up_b2_1, newS1);

    // 3) agg0 = agg_W0 @ new_states1_flat   (320 rows, K=8640)
    gemv_rows<<<SD / 8, 256, 0, stream>>>(
        agg_W0, newS1, agg0raw, SD, KBIG, SD, 0);

    // 4) root MLP + residual -> output (320 floats)
    root_mlp<<<1, 128, 0, stream>>>(
        state0, extin, agg0raw, agg_b0, up_W1_0, up_b1_0, up_W2_0, up_b2_0, out);
}